// FiniteDifferenceGradient_85263690760521
// MI455X (gfx1250) — compile-verified
//
#include <hip/hip_runtime.h>

// Problem geometry: values[4][256][256][256] f32, out[4][3][256][256][256] f32
#define N    256
#define MASK 255u
#define SY   256u
#define SX   65536u        // 256*256
#define SB   16777216u     // 256^3

typedef float v4f __attribute__((ext_vector_type(4)));

__device__ __forceinline__ void async_row_quad(const float* __restrict__ v,
                                               const float* lds_elem,
                                               unsigned byte_off) {
    // GVS mode: mem = SGPR64(base) + VGPR32(byte offset); dest = LDS offset VGPR.
    // Low 32 bits of a generic pointer to __shared__ are the LDS byte offset.
    unsigned lds_off = (unsigned)(uintptr_t)lds_elem;
    asm volatile("global_load_async_to_lds_b128 %0, %1, %2"
                 :: "v"(lds_off), "v"(byte_off), "s"(v)
                 : "memory");
}

__global__ __launch_bounds__(256) void fd_grad_kernel(const float* __restrict__ v,
                                                      float* __restrict__ out) {
    // 6 rows of 256 floats: y0-1 .. y0+4 (periodic)
    __shared__ __align__(16) float lds[6 * N];

    const unsigned t   = threadIdx.x;
    const unsigned bid = blockIdx.x;
    const unsigned y0  = (bid & 63u) << 2;      // 4-row y tile
    const unsigned x   = (bid >> 6) & MASK;
    const unsigned b   = bid >> 14;

    const unsigned baseBX = b * SB + x * SX;    // element offset of (b, x, 0, 0)

    // ---- Phase 1: stage 6 rows into LDS via async DMA (384 float4 quads) ----
    {
        unsigned i  = t;                         // first quad: 0..255
        unsigned r  = i >> 6;                    // lds row 0..5
        unsigned qq = i & 63u;                   // float4 index in row
        unsigned y  = (y0 + r + (N - 1)) & MASK; // y0 - 1 + r, periodic
        async_row_quad(v, &lds[r * N + (qq << 2)],
                       (baseBX + y * SY + (qq << 2)) << 2);
        if (t < 128u) {                          // second quad: 256..383
            i  = t + 256u;
            r  = i >> 6;
            qq = i & 63u;
            y  = (y0 + r + (N - 1)) & MASK;
            async_row_quad(v, &lds[r * N + (qq << 2)],
                           (baseBX + y * SY + (qq << 2)) << 2);
        }
    }

    // ---- Overlap: x-neighbor plane loads (usually L2 hits) ----
    const unsigned rr = t >> 6;                  // row within tile: 0..3
    const unsigned q  = t & 63u;
    const unsigned z0 = q << 2;
    const unsigned y  = y0 + rr;                 // y0 <= 252, no wrap needed

    const v4f xp = *(const v4f*)(v + (size_t)b * SB + (size_t)((x + 1) & MASK) * SX
                                 + y * SY + z0);
    const v4f xm = *(const v4f*)(v + (size_t)b * SB + (size_t)((x + MASK) & MASK) * SX
                                 + y * SY + z0);

    // ---- Wait for this wave's async copies, then block-wide barrier ----
    asm volatile("s_wait_asynccnt 0x0" ::: "memory");
    __syncthreads();

    // ---- Phase 2: neighbors from LDS ----
    const unsigned crow = (rr + 1) * N;
    const v4f   c  = *(const v4f*)&lds[crow + z0];
    const float zm = lds[crow + ((z0 + (N - 1)) & MASK)];
    const float zp = lds[crow + ((z0 + 4u) & MASK)];
    const v4f   ym = *(const v4f*)&lds[rr * N + z0];
    const v4f   yp = *(const v4f*)&lds[(rr + 2) * N + z0];

    const float h = 0.5f;  // 1/(2*spacing), spacing = 1.0
    v4f gx = (xp - xm) * h;
    v4f gy = (yp - ym) * h;
    v4f gz;
    gz.x = (c.y - zm) * h;
    gz.y = (c.z - c.x) * h;
    gz.z = (c.w - c.y) * h;
    gz.w = (zp  - c.z) * h;

    // ---- Non-temporal write-once output: out[b][comp][x][y][z] ----
    float* o = out + (size_t)b * 3u * SB + (size_t)x * SX + (size_t)y * SY + z0;
    __builtin_nontemporal_store(gx, (v4f*)(o));
    __builtin_nontemporal_store(gy, (v4f*)(o + (size_t)SB));
    __builtin_nontemporal_store(gz, (v4f*)(o + 2u * (size_t)SB));
}

extern "C" void kernel_launch(void* const* d_in, const int* in_sizes, int n_in,
                              void* d_out, int out_size, void* d_ws, size_t ws_size,
                              hipStream_t stream) {
    (void)in_sizes; (void)n_in; (void)out_size; (void)d_ws; (void)ws_size;
    const float* v = (const float*)d_in[0];
    float* out     = (float*)d_out;

    // 4 batches * 256 x-planes * 64 y-tiles = 65536 blocks of 256 threads (8 waves)
    dim3 grid(4u * 256u * 64u);
    dim3 block(256u);
    hipLaunchKernelGGL(fd_grad_kernel, grid, block, 0, stream, v, out);
}